// CosineAttention_670014898407
// MI455X (gfx1250) — compile-verified
//
#include <hip/hip_runtime.h>
#include <math.h>

typedef float v2f __attribute__((ext_vector_type(2)));
typedef float v8f __attribute__((ext_vector_type(8)));

#define N_ROWS 8192
#define DDIM   1024
#define EPS_F  1e-8f

// ---------------------------------------------------------------------------
// Kernel 1: invn[row] = 1 / max(||x_row||_2, eps)
// One 256-thread block per row; float4 loads (256*4 = 1024 = DDIM), LDS reduce.
// ---------------------------------------------------------------------------
__global__ __launch_bounds__(256)
void cosattn_rownorm(const float* __restrict__ x, float* __restrict__ invn) {
    __shared__ float red[256];
    const int row = blockIdx.x;
    const int tid = threadIdx.x;
    const float4* xr = (const float4*)(x + (size_t)row * DDIM);
    float4 v = xr[tid];
    float s = v.x * v.x + v.y * v.y + v.z * v.z + v.w * v.w;
    red[tid] = s;
    __syncthreads();
    for (int off = 128; off > 0; off >>= 1) {
        if (tid < off) red[tid] += red[tid + off];
        __syncthreads();
    }
    if (tid == 0) invn[row] = 1.0f / fmaxf(sqrtf(red[0]), EPS_F);
}

// ---------------------------------------------------------------------------
// Kernel 2: G[d1,d2] = sum_i invn[i] * x[i,d1] * x[i,d2]   (G = xn^T @ x, DxD)
// One wave per 32x32 tile of G (2x2 grid of 16x16 WMMA accumulators).
// Per K-step (K = i, step 4): 2 A-frags + 2 B-frags -> 4 independent WMMAs.
// Fragment layouts (ISA 7.12.2, wave32):
//   A(16x4)/B(4x16): VGPR v, lane half h -> K = v + 2h, row/col = lane&15
//   C/D(16x16):      VGPR v, lane half h -> M = v + 8h, N = lane&15
// ---------------------------------------------------------------------------
__global__ __launch_bounds__(256)
void cosattn_gram(const float* __restrict__ x, const float* __restrict__ invn,
                  float* __restrict__ G) {
    const int wave = (int)((blockIdx.x * blockDim.x + threadIdx.x) >> 5);
    const int lane = (int)(threadIdx.x & 31u);
    const int mt = wave >> 5;          // 0..31  (d1 tile of 32)
    const int nt = wave & 31;          // 0..31  (d2 tile of 32)
    const int m0 = mt << 5;
    const int n0 = nt << 5;
    const int half = lane >> 4;        // 0 or 1
    const int l = lane & 15;

    v8f c00 = {}, c01 = {}, c10 = {}, c11 = {};

    #pragma unroll 2
    for (int k0 = 0; k0 < N_ROWS; k0 += 4) {
        const int ka = k0 + 2 * half;          // this half-wave covers rows ka, ka+1
        const float in0 = invn[ka];
        const float in1 = invn[ka + 1];
        const size_t r0 = (size_t)ka * DDIM;
        const size_t r1 = (size_t)(ka + 1) * DDIM;
        v2f a0, a1, b0, b1;
        a0.x = in0 * x[r0 + m0 + l];
        a0.y = in1 * x[r1 + m0 + l];
        a1.x = in0 * x[r0 + m0 + 16 + l];
        a1.y = in1 * x[r1 + m0 + 16 + l];
        b0.x = x[r0 + n0 + l];
        b0.y = x[r1 + n0 + l];
        b1.x = x[r0 + n0 + 16 + l];
        b1.y = x[r1 + n0 + 16 + l];
        c00 = __builtin_amdgcn_wmma_f32_16x16x4_f32(false, a0, false, b0, (short)0, c00, false, false);
        c01 = __builtin_amdgcn_wmma_f32_16x16x4_f32(false, a0, false, b1, (short)0, c01, false, false);
        c10 = __builtin_amdgcn_wmma_f32_16x16x4_f32(false, a1, false, b0, (short)0, c10, false, false);
        c11 = __builtin_amdgcn_wmma_f32_16x16x4_f32(false, a1, false, b1, (short)0, c11, false, false);
    }

    const int mrow = m0 + (half << 3);
    #pragma unroll
    for (int v = 0; v < 8; ++v) {
        G[(size_t)(mrow + v) * DDIM + n0 + l]            = c00[v];
        G[(size_t)(mrow + v) * DDIM + n0 + 16 + l]       = c01[v];
        G[(size_t)(mrow + 16 + v) * DDIM + n0 + l]       = c10[v];
        G[(size_t)(mrow + 16 + v) * DDIM + n0 + 16 + l]  = c11[v];
    }
}

// ---------------------------------------------------------------------------
// Kernel 3: out[i,d] = sigmoid( invn[i] * sum_k x[i,k] * G[k,d] )
// One wave per 32x32 output tile; K = DDIM in steps of 4.
// A rows are contiguous in x -> float2 (8B-aligned) loads; 6 loads / 4 WMMAs.
// ---------------------------------------------------------------------------
__global__ __launch_bounds__(256)
void cosattn_out(const float* __restrict__ x, const float* __restrict__ G,
                 const float* __restrict__ invn, float* __restrict__ out) {
    const int wave = (int)((blockIdx.x * blockDim.x + threadIdx.x) >> 5);
    const int lane = (int)(threadIdx.x & 31u);
    const int mt = wave >> 5;          // 0..255 (row tile of 32)
    const int nt = wave & 31;          // 0..31  (col tile of 32)
    const int i0 = mt << 5;
    const int n0 = nt << 5;
    const int half = lane >> 4;
    const int l = lane & 15;

    const float* xrow0 = x + (size_t)(i0 + l) * DDIM;        // rows of A-tile 0
    const float* xrow1 = x + (size_t)(i0 + 16 + l) * DDIM;   // rows of A-tile 1

    v8f c00 = {}, c01 = {}, c10 = {}, c11 = {};

    #pragma unroll 2
    for (int k0 = 0; k0 < DDIM; k0 += 4) {
        const int ka = k0 + 2 * half;
        v2f a0 = *(const v2f*)(xrow0 + ka);    // x[i0+l,    ka..ka+1]
        v2f a1 = *(const v2f*)(xrow1 + ka);    // x[i0+16+l, ka..ka+1]
        const size_t g0 = (size_t)ka * DDIM;
        const size_t g1 = (size_t)(ka + 1) * DDIM;
        v2f b0, b1;
        b0.x = G[g0 + n0 + l];
        b0.y = G[g1 + n0 + l];
        b1.x = G[g0 + n0 + 16 + l];
        b1.y = G[g1 + n0 + 16 + l];
        c00 = __builtin_amdgcn_wmma_f32_16x16x4_f32(false, a0, false, b0, (short)0, c00, false, false);
        c01 = __builtin_amdgcn_wmma_f32_16x16x4_f32(false, a0, false, b1, (short)0, c01, false, false);
        c10 = __builtin_amdgcn_wmma_f32_16x16x4_f32(false, a1, false, b0, (short)0, c10, false, false);
        c11 = __builtin_amdgcn_wmma_f32_16x16x4_f32(false, a1, false, b1, (short)0, c11, false, false);
    }

    const int mrow = i0 + (half << 3);
    #pragma unroll
    for (int v = 0; v < 8; ++v) {
        const int r0 = mrow + v;
        const int r1 = mrow + 16 + v;
        const float s0 = invn[r0];
        const float s1 = invn[r1];
        out[(size_t)r0 * DDIM + n0 + l]      = 1.0f / (1.0f + expf(-s0 * c00[v]));
        out[(size_t)r0 * DDIM + n0 + 16 + l] = 1.0f / (1.0f + expf(-s0 * c01[v]));
        out[(size_t)r1 * DDIM + n0 + l]      = 1.0f / (1.0f + expf(-s1 * c10[v]));
        out[(size_t)r1 * DDIM + n0 + 16 + l] = 1.0f / (1.0f + expf(-s1 * c11[v]));
    }
}

// ---------------------------------------------------------------------------
// Host launcher. d_in[0] = x [8192*1024 f32], d_in[1] = W (unused).
// d_out = attn [8192*1024 f32]. ws: invn (8192 f32) then G (1024*1024 f32).
// ---------------------------------------------------------------------------
extern "C" void kernel_launch(void* const* d_in, const int* in_sizes, int n_in,
                              void* d_out, int out_size, void* d_ws, size_t ws_size,
                              hipStream_t stream) {
    (void)in_sizes; (void)n_in; (void)out_size; (void)ws_size;
    const float* x = (const float*)d_in[0];
    float* out  = (float*)d_out;
    float* ws   = (float*)d_ws;
    float* invn = ws;                  // 8192 floats
    float* G    = ws + N_ROWS;         // 1024*1024 floats (4 MB)

    // 1) row inverse norms
    cosattn_rownorm<<<N_ROWS, 256, 0, stream>>>(x, invn);

    // 2) G = xn^T @ x : 32x32 grid of 32x32 tiles -> 1024 waves -> 128 blocks
    cosattn_gram<<<128, 256, 0, stream>>>(x, invn, G);

    // 3) out = sigmoid(diag(invn) * (x @ G)) : 256x32 tiles -> 8192 waves
    cosattn_out<<<1024, 256, 0, stream>>>(x, G, invn, out);
}